// SparseLinear_21809843929266
// MI455X (gfx1250) — compile-verified
//
#include <hip/hip_runtime.h>
#include <hip/hip_bf16.h>

#define B_      512
#define IN_     65536
#define OUT_    16384
#define K_      32
#define THREADS 1024   // 32 waves (wave32) per workgroup

typedef int v4i __attribute__((ext_vector_type(4)));

#if defined(__gfx1250__) && __has_builtin(__builtin_amdgcn_global_load_async_to_lds_b128)
#define HAVE_ASYNC_LDS 1
#else
#define HAVE_ASYNC_LDS 0
#endif

__global__ __launch_bounds__(THREADS)
void sparse_linear_rowlds(const float* __restrict__ x,
                          const int*   __restrict__ indices,
                          const float* __restrict__ weight,
                          const float* __restrict__ bias,
                          float*       __restrict__ out)
{
    extern __shared__ float xs[];          // 65536 floats = 256 KB (<= 320 KB WGP LDS)
    const int b   = blockIdx.x;            // one workgroup per batch row
    const int tid = threadIdx.x;
    const float* __restrict__ xrow = x + (size_t)b * IN_;

    // ---- Stage 1: stream the whole 256 KB row into LDS ----
#if HAVE_ASYNC_LDS
    // gfx1250 async copy: global -> LDS without a VGPR round-trip,
    // tracked by ASYNCcnt. 16 b128 transfers per thread, fully coalesced.
    for (int j = tid; j < IN_ / 4; j += THREADS) {
        __builtin_amdgcn_global_load_async_to_lds_b128(
            (v4i*)(xrow + 4 * j),     // global source (generic ptr, AS inferred)
            (v4i*)(xs + 4 * j),       // LDS destination
            /*offset=*/0, /*cpol=*/0);
    }
#if __has_builtin(__builtin_amdgcn_s_wait_asynccnt)
    __builtin_amdgcn_s_wait_asynccnt(0);
#else
    asm volatile("s_wait_asynccnt 0" ::: "memory");
#endif
#else
    // Fallback: vector copy through VGPRs.
    {
        const float4* __restrict__ src = (const float4*)xrow;
        float4*       __restrict__ dst = (float4*)xs;
        for (int j = tid; j < IN_ / 4; j += THREADS) dst[j] = src[j];
    }
#endif
    __syncthreads();

    // ---- Stage 2: each thread produces OUT_/THREADS = 16 outputs ----
    // indices/weight rows are 128 B each -> b128 loads, and the whole 4 MB of
    // idx+w stays resident in the 192 MB L2 across all 512 row-workgroups.
    const size_t out_base = (size_t)b * OUT_;
    for (int o = tid; o < OUT_; o += THREADS) {
        const int4*   __restrict__ i4 = (const int4*)  (indices + (size_t)o * K_);
        const float4* __restrict__ w4 = (const float4*)(weight  + (size_t)o * K_);
        float acc = 0.0f;
#pragma unroll
        for (int j = 0; j < K_ / 4; ++j) {
            const int4   iv = i4[j];
            const float4 wv = w4[j];
            acc = fmaf(xs[iv.x], wv.x, acc);   // ds_load_b32 gathers
            acc = fmaf(xs[iv.y], wv.y, acc);
            acc = fmaf(xs[iv.z], wv.z, acc);
            acc = fmaf(xs[iv.w], wv.w, acc);
        }
        out[out_base + o] = acc + bias[o];     // coalesced b32 store
    }
}

extern "C" void kernel_launch(void* const* d_in, const int* in_sizes, int n_in,
                              void* d_out, int out_size, void* d_ws, size_t ws_size,
                              hipStream_t stream)
{
    const float* x       = (const float*)d_in[0];
    const int*   indices = (const int*)  d_in[1];
    const float* weight  = (const float*)d_in[2];
    const float* bias    = (const float*)d_in[3];
    float*       out     = (float*)d_out;

    const size_t shmem = (size_t)IN_ * sizeof(float);   // 262144 bytes
    hipLaunchKernelGGL(sparse_linear_rowlds,
                       dim3(B_), dim3(THREADS), shmem, stream,
                       x, indices, weight, bias, out);
}